// SimpleGCN_18330920419812
// MI455X (gfx1250) — compile-verified
//
#include <hip/hip_runtime.h>
#include <hip/hip_bf16.h>

typedef __attribute__((ext_vector_type(2))) float v2f;
typedef __attribute__((ext_vector_type(8))) float v8f;

#define F_IN 128
#define H1C  16
#define H2C  8
#define NC   10
#define NPG  1024

// ---------------------------------------------------------------------------
// h1[N,16] = x[N,128] @ W1[128,16]  via v_wmma_f32_16x16x4_f32 (exact fp32)
// One wave per 16-row tile; 32 WMMA steps over K=128.
// ---------------------------------------------------------------------------
__global__ void gemm1_wmma(const float* __restrict__ x,
                           const float* __restrict__ W1,
                           float* __restrict__ h1, int n_nodes) {
  int wave = (int)((blockIdx.x * blockDim.x + threadIdx.x) >> 5);
  int lane = threadIdx.x & 31;
  int base = wave * 16;
  if (base >= n_nodes) return;
  int m    = lane & 15;            // A-row / B-col for this lane
  int koff = (lane >> 4) << 1;     // lanes 16-31 hold K=k+2,k+3
  v8f acc = {};
  const float* xrow = x + (size_t)(base + m) * F_IN;
#pragma unroll
  for (int k0 = 0; k0 < F_IN; k0 += 4) {
    v2f a, b;
    a.x = xrow[k0 + koff];
    a.y = xrow[k0 + koff + 1];
    b.x = W1[(k0 + koff) * H1C + m];
    b.y = W1[(k0 + koff + 1) * H1C + m];
    acc = __builtin_amdgcn_wmma_f32_16x16x4_f32(false, a, false, b,
                                                (short)0, acc, false, false);
  }
  int col    = lane & 15;
  int rowoff = (lane >> 4) << 3;   // lanes 16-31 hold rows M=8..15
#pragma unroll
  for (int v = 0; v < 8; ++v)
    h1[(size_t)(base + rowoff + v) * H1C + col] = acc[v];
}

// ---------------------------------------------------------------------------
// h2[N,8] = h1a[N,16] @ W2[16,8] (W2 zero-padded to 16 cols) via WMMA f32
// ---------------------------------------------------------------------------
__global__ void gemm2_wmma(const float* __restrict__ h1a,
                           const float* __restrict__ W2,
                           float* __restrict__ h2, int n_nodes) {
  int wave = (int)((blockIdx.x * blockDim.x + threadIdx.x) >> 5);
  int lane = threadIdx.x & 31;
  int base = wave * 16;
  if (base >= n_nodes) return;
  int m    = lane & 15;
  int koff = (lane >> 4) << 1;
  v8f acc = {};
  const float* arow = h1a + (size_t)(base + m) * H1C;
#pragma unroll
  for (int k0 = 0; k0 < H1C; k0 += 4) {
    v2f a, b;
    a.x = arow[k0 + koff];
    a.y = arow[k0 + koff + 1];
    b.x = (m < H2C) ? W2[(k0 + koff) * H2C + m] : 0.0f;
    b.y = (m < H2C) ? W2[(k0 + koff + 1) * H2C + m] : 0.0f;
    acc = __builtin_amdgcn_wmma_f32_16x16x4_f32(false, a, false, b,
                                                (short)0, acc, false, false);
  }
  int col    = lane & 15;
  int rowoff = (lane >> 4) << 3;
  if (col < H2C) {
#pragma unroll
    for (int v = 0; v < 8; ++v)
      h2[(size_t)(base + rowoff + v) * H2C + col] = acc[v];
  }
}

// ---------------------------------------------------------------------------
__global__ void init_deg(float* __restrict__ deg, int n) {
  int i = blockIdx.x * blockDim.x + threadIdx.x;
  if (i < n) deg[i] = 1.0f;                         // self-loop
}

__global__ void count_deg(const int* __restrict__ dst, float* __restrict__ deg, int e) {
  int i = blockIdx.x * blockDim.x + threadIdx.x;
  if (i < e) atomicAdd(&deg[dst[i]], 1.0f);
}

// dinv = rsqrt(deg); layer-1 self-loop message agg1[i] = h1[i] * dinv^2
__global__ void prep_self1(const float* __restrict__ deg, float* __restrict__ dinv,
                           const float* __restrict__ h1, float* __restrict__ agg1, int n) {
  int gid = blockIdx.x * blockDim.x + threadIdx.x;
  if (gid >= n * H1C) return;
  int i = gid >> 4, f = gid & 15;
  float di = rsqrtf(deg[i]);
  if (f == 0) dinv[i] = di;
  agg1[(size_t)i * H1C + f] = h1[(size_t)i * H1C + f] * (di * di);
}

// 16 threads per edge: coalesced gather of h1[src][0:16] + contiguous atomics
__global__ void agg1_edges(const int* __restrict__ src, const int* __restrict__ dst,
                           const float* __restrict__ dinv, const float* __restrict__ h1,
                           float* __restrict__ agg1, int e) {
  long long gid = (long long)blockIdx.x * blockDim.x + threadIdx.x;
  if (gid >= (long long)e * H1C) return;
  int ei = (int)(gid >> 4), f = (int)(gid & 15);
  int s = src[ei], d = dst[ei];
  float norm = dinv[s] * dinv[d];
  atomicAdd(&agg1[(size_t)d * H1C + f], h1[(size_t)s * H1C + f] * norm);
}

__global__ void bias_relu1(float* __restrict__ a, const float* __restrict__ b, int n) {
  int gid = blockIdx.x * blockDim.x + threadIdx.x;
  if (gid >= n * H1C) return;
  float v = a[gid] + b[gid & 15];
  a[gid] = v > 0.0f ? v : 0.0f;
}

__global__ void prep_self2(const float* __restrict__ dinv, const float* __restrict__ h2,
                           float* __restrict__ agg2, int n) {
  int gid = blockIdx.x * blockDim.x + threadIdx.x;
  if (gid >= n * H2C) return;
  int i = gid >> 3;
  float di = dinv[i];
  agg2[gid] = h2[gid] * (di * di);
}

// 8 threads per edge
__global__ void agg2_edges(const int* __restrict__ src, const int* __restrict__ dst,
                           const float* __restrict__ dinv, const float* __restrict__ h2,
                           float* __restrict__ agg2, int e) {
  long long gid = (long long)blockIdx.x * blockDim.x + threadIdx.x;
  if (gid >= (long long)e * H2C) return;
  int ei = (int)(gid >> 3), f = (int)(gid & 7);
  int s = src[ei], d = dst[ei];
  float norm = dinv[s] * dinv[d];
  atomicAdd(&agg2[(size_t)d * H2C + f], h2[(size_t)s * H2C + f] * norm);
}

__global__ void bias_relu2(float* __restrict__ a, const float* __restrict__ b, int n) {
  int gid = blockIdx.x * blockDim.x + threadIdx.x;
  if (gid >= n * H2C) return;
  float v = a[gid] + b[gid & 7];
  a[gid] = v > 0.0f ? v : 0.0f;
}

// out[g,c] = sum_j h2a[g*8192+j] * fc_w[c*8192+j] + fc_b[c]; block per graph
__global__ void fc_head(const float* __restrict__ h, const float* __restrict__ fcw,
                        const float* __restrict__ fcb, float* __restrict__ out) {
  __shared__ float red[256];
  int g = blockIdx.x;
  const float* hg = h + (size_t)g * (NPG * H2C);
  for (int c = 0; c < NC; ++c) {
    const float* w = fcw + (size_t)c * (NPG * H2C);
    float s = 0.0f;
    for (int j = threadIdx.x; j < NPG * H2C; j += blockDim.x) s += hg[j] * w[j];
    red[threadIdx.x] = s;
    __syncthreads();
    for (int off = 128; off > 0; off >>= 1) {
      if ((int)threadIdx.x < off) red[threadIdx.x] += red[threadIdx.x + off];
      __syncthreads();
    }
    if (threadIdx.x == 0) out[g * NC + c] = red[0] + fcb[c];
    __syncthreads();
  }
}

// ---------------------------------------------------------------------------
extern "C" void kernel_launch(void* const* d_in, const int* in_sizes, int n_in,
                              void* d_out, int out_size, void* d_ws, size_t ws_size,
                              hipStream_t stream) {
  const float* x    = (const float*)d_in[0];
  const int*   ei   = (const int*)d_in[1];
  const float* W1   = (const float*)d_in[2];
  const float* b1   = (const float*)d_in[3];
  const float* W2   = (const float*)d_in[4];
  const float* b2   = (const float*)d_in[5];
  const float* fcw  = (const float*)d_in[6];
  const float* fcb  = (const float*)d_in[7];
  float* out = (float*)d_out;

  const int N = in_sizes[0] / F_IN;       // 65536
  const int E = in_sizes[1] / 2;          // 2097152
  const int NG = N / NPG;                 // 64
  const int* src = ei;
  const int* dst = ei + E;

  float* ws   = (float*)d_ws;
  float* h1   = ws;                       // N*16
  float* agg1 = h1   + (size_t)N * H1C;   // N*16  (becomes h1a after relu)
  float* deg  = agg1 + (size_t)N * H1C;   // N
  float* dinv = deg  + N;                 // N
  float* h2   = dinv + N;                 // N*8
  float* agg2 = h2   + (size_t)N * H2C;   // N*8   (becomes h2a after relu)

  const int T = 256;
  auto cdiv = [](long long a, long long b) { return (int)((a + b - 1) / b); };

  init_deg  <<<cdiv(N, T), T, 0, stream>>>(deg, N);
  gemm1_wmma<<<cdiv((long long)(N / 16) * 32, T), T, 0, stream>>>(x, W1, h1, N);
  count_deg <<<cdiv(E, T), T, 0, stream>>>(dst, deg, E);
  prep_self1<<<cdiv((long long)N * H1C, T), T, 0, stream>>>(deg, dinv, h1, agg1, N);
  agg1_edges<<<cdiv((long long)E * H1C, T), T, 0, stream>>>(src, dst, dinv, h1, agg1, E);
  bias_relu1<<<cdiv((long long)N * H1C, T), T, 0, stream>>>(agg1, b1, N);
  gemm2_wmma<<<cdiv((long long)(N / 16) * 32, T), T, 0, stream>>>(agg1, W2, h2, N);
  prep_self2<<<cdiv((long long)N * H2C, T), T, 0, stream>>>(dinv, h2, agg2, N);
  agg2_edges<<<cdiv((long long)E * H2C, T), T, 0, stream>>>(src, dst, dinv, h2, agg2, E);
  bias_relu2<<<cdiv((long long)N * H2C, T), T, 0, stream>>>(agg2, b2, N);
  fc_head   <<<NG, T, 0, stream>>>(agg2, fcw, fcb, out);
}